// Dendrite_layer_84782654423696
// MI455X (gfx1250) — compile-verified
//
#include <hip/hip_runtime.h>
#include <stdint.h>

#define PI_F   3.14159265358979323846f
#define DIM_H  512
#define DIM_W  512
#define NPIX   (DIM_H * DIM_W)
#define NSYN   8
#define NBATCH 4
#define TPB    256

// ---------------------------------------------------------------------------
// CDNA5 async global->LDS copy (ASYNCcnt-tracked), GVS mode:
//   mem_addr = SADDR(64b sgpr pair) + VADDR(32b per-lane byte offset)
//   LDS dest = per-lane 32-bit LDS byte address in the VDST vgpr.
// ---------------------------------------------------------------------------
__device__ __forceinline__ void async_ld_f32(uint32_t lds_byte_addr,
                                             const float* base,
                                             uint32_t byte_off) {
    asm volatile("global_load_async_to_lds_b32 %0, %1, %2"
                 :: "v"(lds_byte_addr), "v"(byte_off), "s"(base)
                 : "memory");
}

__device__ __forceinline__ void wait_async_le3() {
    asm volatile("s_wait_asynccnt 3" ::: "memory");
}
__device__ __forceinline__ void wait_async_0() {
    asm volatile("s_wait_asynccnt 0" ::: "memory");
}

// to_decard for size==512 (both dims): (2*f/pi + 1) * 0.5 * 511
__device__ __forceinline__ float to_decard512(float angle) {
    float s = __sinf(angle);
    float c = __cosf(angle);
    float f = asinf(s) * c * __frsqrt_rn(1.0f - s * s + 1e-6f);
    return (f * (2.0f / PI_F) + 1.0f) * 0.5f * 511.0f;
}

__global__ __launch_bounds__(TPB) void dendrite_kernel(
    const float* __restrict__ signal,   // (4, 512, 512)
    const float* __restrict__ weight,   // (8, 512, 512)
    const float* __restrict__ coord0,   // (8, 512, 512)
    const float* __restrict__ coord1,   // (8, 512, 512)
    float* __restrict__ out)            // (4, 512, 512)
{
    // double-buffered staging: [stage 0/1][c0,c1,wt][lane]
    __shared__ float sbuf[2][3][TPB];

    const int tid = threadIdx.x;
    const int p   = blockIdx.x * TPB + tid;     // pixel = h*512 + w

    // Low 32 bits of a generic shared pointer == LDS byte offset (aperture
    // form is {shared_base_hi32, lds_offset32}); that is what the async
    // instruction's VDST operand expects.
    const uint32_t lds0 = (uint32_t)(uintptr_t)(&sbuf[0][0][0]);
    const uint32_t lane_b = (uint32_t)tid * 4u;
    #define LDS_ADDR(buf, arr) (lds0 + (uint32_t)(((buf) * 3 + (arr)) * TPB) * 4u + lane_b)

    const float STEP = PI_F / 511.0f;   // PI * delta / (IN_SIZE[d]-1)

    const float* sig0 = signal;
    const float* sig1 = signal + NPIX;
    const float* sig2 = signal + 2 * NPIX;
    const float* sig3 = signal + 3 * NPIX;

    // prefetch stage for synapse 0
    {
        const uint32_t off = (uint32_t)p * 4u;
        async_ld_f32(LDS_ADDR(0, 0), coord0, off);
        async_ld_f32(LDS_ADDR(0, 1), coord1, off);
        async_ld_f32(LDS_ADDR(0, 2), weight, off);
    }

    float acc0 = 0.f, acc1 = 0.f, acc2 = 0.f, acc3 = 0.f;

    for (int s = 0; s < NSYN; ++s) {
        // software pipeline: kick off synapse s+1 into the other buffer,
        // then wait until stage s (the oldest 3 async ops) has landed.
        if (s + 1 < NSYN) {
            const uint32_t off = (uint32_t)((s + 1) * NPIX + p) * 4u;
            const int nb = (s + 1) & 1;
            async_ld_f32(LDS_ADDR(nb, 0), coord0, off);
            async_ld_f32(LDS_ADDR(nb, 1), coord1, off);
            async_ld_f32(LDS_ADDR(nb, 2), weight, off);
            wait_async_le3();
        } else {
            wait_async_0();
        }

        const int cb = s & 1;
        const float c0 = sbuf[cb][0][tid];
        const float c1 = sbuf[cb][1][tid];
        const float wt = sbuf[cb][2][tid];

        const float dec0 = to_decard512(c0);
        const float dec1 = to_decard512(c1);

        float dx2[3], dy2[3];
        int   i0[3], i1[3];
        #pragma unroll
        for (int j = 0; j < 3; ++j) {
            const float dlt = (float)(j - 1) * STEP;
            // reference: src uses the UNclamped rounded index; gather clamps.
            const float r0 = rintf(to_decard512(c0 + dlt));   // round half->even
            const float r1 = rintf(to_decard512(c1 + dlt));
            const float d0 = r0 - dec0;
            const float d1 = r1 - dec1;
            dx2[j] = d0 * d0;
            dy2[j] = d1 * d1;
            int q0 = (int)r0; q0 = q0 < 0 ? 0 : (q0 > DIM_H - 1 ? DIM_H - 1 : q0);
            int q1 = (int)r1; q1 = q1 < 0 ? 0 : (q1 > DIM_W - 1 ? DIM_W - 1 : q1);
            i0[j] = q0;
            i1[j] = q1;
        }

        #pragma unroll
        for (int j0 = 0; j0 < 3; ++j0) {
            const int rowoff = i0[j0] * DIM_W;
            #pragma unroll
            for (int j1 = 0; j1 < 3; ++j1) {
                const float srcv = __fsqrt_rn(dx2[j0] + dy2[j1]);
                // sigmoid(6*(1 - 2*src)) = 1 / (1 + exp(12*src - 6))
                const float mask = 1.0f / (1.0f + __expf(12.0f * srcv - 6.0f));
                const float wm   = wt * mask;
                const int   idx  = rowoff + i1[j1];
                acc0 = fmaf(sig0[idx], wm, acc0);
                acc1 = fmaf(sig1[idx], wm, acc1);
                acc2 = fmaf(sig2[idx], wm, acc2);
                acc3 = fmaf(sig3[idx], wm, acc3);
            }
        }
    }

    out[p]            = acc0;
    out[NPIX + p]     = acc1;
    out[2 * NPIX + p] = acc2;
    out[3 * NPIX + p] = acc3;
    #undef LDS_ADDR
}

extern "C" void kernel_launch(void* const* d_in, const int* in_sizes, int n_in,
                              void* d_out, int out_size, void* d_ws, size_t ws_size,
                              hipStream_t stream) {
    (void)in_sizes; (void)n_in; (void)out_size; (void)d_ws; (void)ws_size;
    const float* signal = (const float*)d_in[0];
    const float* weight = (const float*)d_in[1];
    const float* coord0 = (const float*)d_in[2];
    const float* coord1 = (const float*)d_in[3];
    float* out = (float*)d_out;

    dim3 grid(NPIX / TPB);   // 1024 blocks
    dim3 block(TPB);         // 8 wave32s per block -> EXEC all-ones per lane op
    hipLaunchKernelGGL(dendrite_kernel, grid, block, 0, stream,
                       signal, weight, coord0, coord1, out);
}